// GCNLayer_58308476010684
// MI455X (gfx1250) — compile-verified
//
#include <hip/hip_runtime.h>

typedef __attribute__((ext_vector_type(2))) float v2f;
typedef __attribute__((ext_vector_type(8))) float v8f;

#define IN_FEATS   128
#define OUT_FEATS  128
#define ROWS_PER_BLOCK 80
#define TILES_PER_BLOCK (ROWS_PER_BLOCK / 16)

// ---------------------------------------------------------------------------
// Kernel 1: zero agg [N,128] and deg [N] (contiguous in workspace)
// ---------------------------------------------------------------------------
__global__ void gcn_zero_kernel(float* __restrict__ p, int n) {
    int i = blockIdx.x * blockDim.x + threadIdx.x;
    if (i < n) p[i] = 0.0f;
}

// ---------------------------------------------------------------------------
// Kernel 2: edge scatter. One wave32 per edge, float4 per lane.
// agg[dst] += feat[src]; deg[dst] += 1. All operands L2-resident (51 MB each,
// 192 MB L2) -> native f32 atomics at L2, no HBM round trips.
// ---------------------------------------------------------------------------
__global__ void gcn_scatter_kernel(const float* __restrict__ feat,
                                   const int*   __restrict__ src,
                                   const int*   __restrict__ dst,
                                   float* __restrict__ agg,
                                   float* __restrict__ deg,
                                   int n_edges) {
    int gid  = blockIdx.x * blockDim.x + threadIdx.x;
    int edge = gid >> 5;
    int lane = gid & 31;
    if (edge >= n_edges) return;

    int s = src[edge];
    int d = dst[edge];

    const float4 v = ((const float4*)(feat + (size_t)s * IN_FEATS))[lane];
    float* ap = agg + (size_t)d * IN_FEATS + lane * 4;
    __hip_atomic_fetch_add(ap + 0, v.x, __ATOMIC_RELAXED, __HIP_MEMORY_SCOPE_AGENT);
    __hip_atomic_fetch_add(ap + 1, v.y, __ATOMIC_RELAXED, __HIP_MEMORY_SCOPE_AGENT);
    __hip_atomic_fetch_add(ap + 2, v.z, __ATOMIC_RELAXED, __HIP_MEMORY_SCOPE_AGENT);
    __hip_atomic_fetch_add(ap + 3, v.w, __ATOMIC_RELAXED, __HIP_MEMORY_SCOPE_AGENT);
    if (lane == 0)
        __hip_atomic_fetch_add(deg + d, 1.0f, __ATOMIC_RELAXED, __HIP_MEMORY_SCOPE_AGENT);
}

// ---------------------------------------------------------------------------
// Kernel 3: fused dual GEMM + epilogue using V_WMMA_F32_16X16X4_F32.
//   out = relu(feat@Wv + norm * (agg@Wu) + bias)
// 256 threads = 8 waves; wave w owns output columns [16w, 16w+16).
// Both 128x128 f32 weights staged once into 128 KB LDS, reused across 5 row
// tiles (80 rows/block). A-fragments read from global (WGP$ catches the 8x
// inter-wave reuse); next tile's rows are prefetched (global_prefetch_b8).
// ---------------------------------------------------------------------------
__global__ void __launch_bounds__(256, 2)
gcn_gemm_kernel(const float* __restrict__ feat,
                const float* __restrict__ agg,
                const float* __restrict__ wu,
                const float* __restrict__ wv,
                const float* __restrict__ bias,
                const float* __restrict__ deg,
                float* __restrict__ out) {
    extern __shared__ float smem[];
    float* lds_u = smem;                         // 128*128 f32
    float* lds_v = smem + IN_FEATS * OUT_FEATS;  // 128*128 f32

    // Cooperative stage of both weight matrices into LDS (float4 = b128 path).
    const float4* wu4 = (const float4*)wu;
    const float4* wv4 = (const float4*)wv;
    float4* lu4 = (float4*)lds_u;
    float4* lv4 = (float4*)lds_v;
    for (int i = threadIdx.x; i < IN_FEATS * OUT_FEATS / 4; i += blockDim.x) {
        lu4[i] = wu4[i];
        lv4[i] = wv4[i];
    }
    __syncthreads();

    const int lane  = threadIdx.x & 31;
    const int wave  = threadIdx.x >> 5;
    const int n0    = wave * 16;          // column tile of this wave
    const int mn    = lane & 15;          // row (A) / col (B,C) within tile
    const int khalf = (lane >> 4) << 1;   // lanes 0-15 -> K+0,K+1 ; 16-31 -> K+2,K+3

    const float bn = bias[n0 + mn];

    for (int t = 0; t < TILES_PER_BLOCK; ++t) {
        const int row0 = blockIdx.x * ROWS_PER_BLOCK + t * 16;

        const float* fa = feat + (size_t)(row0 + mn) * IN_FEATS + khalf;
        const float* ga = agg  + (size_t)(row0 + mn) * IN_FEATS + khalf;

        // Prefetch next tile's A rows (speculative; dropped past grid end).
        __builtin_prefetch(fa + 16 * IN_FEATS, 0, 1);
        __builtin_prefetch(ga + 16 * IN_FEATS, 0, 1);

        v8f acc_v = {};
        v8f acc_u = {};

#pragma unroll
        for (int k = 0; k < IN_FEATS; k += 4) {
            // A fragments (16x4 f32, 2 VGPRs/lane): 8-byte aligned b64 loads.
            v2f a_f = *(const v2f*)(fa + k);
            v2f a_g = *(const v2f*)(ga + k);

            // B fragments (4x16 f32, 2 VGPRs/lane) from LDS-resident weights.
            const int kb = k + khalf;
            v2f b_v, b_u;
            b_v.x = lds_v[(kb + 0) * OUT_FEATS + n0 + mn];
            b_v.y = lds_v[(kb + 1) * OUT_FEATS + n0 + mn];
            b_u.x = lds_u[(kb + 0) * OUT_FEATS + n0 + mn];
            b_u.y = lds_u[(kb + 1) * OUT_FEATS + n0 + mn];

            acc_v = __builtin_amdgcn_wmma_f32_16x16x4_f32(
                false, a_f, false, b_v, (short)0, acc_v, false, false);
            acc_u = __builtin_amdgcn_wmma_f32_16x16x4_f32(
                false, a_g, false, b_u, (short)0, acc_u, false, false);
        }

        // Epilogue. C/D layout: VGPR i holds row (i) for lanes 0-15, (8+i)
        // for lanes 16-31; column = n0 + mn. Pull this lane-half's 8 degree
        // values with two b128 loads, then one v_rcp_f32 per row (1-ulp,
        // vs. the ~10-op IEEE divide expansion).
        const int mbase = (lane < 16) ? 0 : 8;
        const float4 dg0 = *(const float4*)(deg + row0 + mbase);
        const float4 dg1 = *(const float4*)(deg + row0 + mbase + 4);
        float nrm[8];
        nrm[0] = __builtin_amdgcn_rcpf(fmaxf(dg0.x, 1.0f));
        nrm[1] = __builtin_amdgcn_rcpf(fmaxf(dg0.y, 1.0f));
        nrm[2] = __builtin_amdgcn_rcpf(fmaxf(dg0.z, 1.0f));
        nrm[3] = __builtin_amdgcn_rcpf(fmaxf(dg0.w, 1.0f));
        nrm[4] = __builtin_amdgcn_rcpf(fmaxf(dg1.x, 1.0f));
        nrm[5] = __builtin_amdgcn_rcpf(fmaxf(dg1.y, 1.0f));
        nrm[6] = __builtin_amdgcn_rcpf(fmaxf(dg1.z, 1.0f));
        nrm[7] = __builtin_amdgcn_rcpf(fmaxf(dg1.w, 1.0f));

#pragma unroll
        for (int i = 0; i < 8; ++i) {
            const int mm = mbase + i;
            const float val = acc_v[i] + nrm[i] * acc_u[i] + bn;
            out[(size_t)(row0 + mm) * OUT_FEATS + n0 + mn] = fmaxf(val, 0.0f);
        }
    }
}

// ---------------------------------------------------------------------------
extern "C" void kernel_launch(void* const* d_in, const int* in_sizes, int n_in,
                              void* d_out, int out_size, void* d_ws, size_t ws_size,
                              hipStream_t stream) {
    (void)n_in; (void)out_size; (void)ws_size;

    const float* feat = (const float*)d_in[0];
    const float* wu   = (const float*)d_in[1];
    const float* wv   = (const float*)d_in[2];
    const float* bias = (const float*)d_in[3];
    const int*   src  = (const int*)d_in[4];
    const int*   dst  = (const int*)d_in[5];
    float* out = (float*)d_out;

    const int n_nodes = in_sizes[0] / IN_FEATS;   // 100000
    const int n_edges = in_sizes[4];              // 1600000

    float* agg = (float*)d_ws;                    // [n_nodes, 128]
    float* deg = agg + (size_t)n_nodes * IN_FEATS;// [n_nodes]

    // 1) zero agg + deg (contiguous)
    const int zn = n_nodes * (IN_FEATS + 1);
    gcn_zero_kernel<<<(zn + 255) / 256, 256, 0, stream>>>(agg, zn);

    // 2) edge scatter: one wave per edge
    const long sthreads = (long)n_edges * 32;
    gcn_scatter_kernel<<<(int)((sthreads + 255) / 256), 256, 0, stream>>>(
        feat, src, dst, agg, deg, n_edges);

    // 3) fused dual WMMA GEMM + normalization + bias + relu
    const int blocks = n_nodes / ROWS_PER_BLOCK;  // 1250
    const size_t shmem = 2u * IN_FEATS * OUT_FEATS * sizeof(float); // 128 KB
    gcn_gemm_kernel<<<blocks, 256, shmem, stream>>>(
        feat, agg, wu, wv, bias, deg, out);
}